// CorrelatedSphericalField_44255343018727
// MI455X (gfx1250) — compile-verified
//
#include <hip/hip_runtime.h>

// ---------------------------------------------------------------------------
// CorrelatedSphericalField on MI455X (gfx1250, wave32, WMMA).
//
//   1) ar1_scan      : AR(1), T=8  -> Abuf[m][re/im][l][row]   (K-major)
//   2) legendre_gemm : per-m fp32 WMMA GEMM (128x361)*(361x361) -> Xbuf[c][row]
//   3) build_w       : 4pi-weighted cos/-sin DFT matrix (736 x 722)
//   4) idft_gemm     : fp32 WMMA GEMM (46208 x 736)*(736 x 722) -> out
//
// Full-fp32 matrix path (V_WMMA_F32_16X16X4_F32): reference math is fp32 and
// high-degree Legendre values have too much dynamic range for bf16/f16.
// Each wave computes a 32x32 macro-tile (2x2 WMMA tiles): A/B fragments are
// reused across the 2x2 -> 2 loads per WMMA instead of 4. Tile ids are
// readfirstlane'd so K-loops are scalar (EXEC all-ones across every WMMA);
// in-loop guards are index clamps, never EXEC-modifying branches.
// ---------------------------------------------------------------------------

typedef __attribute__((ext_vector_type(2))) float v2f;
typedef __attribute__((ext_vector_type(4))) float v4f;
typedef __attribute__((ext_vector_type(8))) float v8f;

#define TT     8      // time steps
#define NLn    16     // n_latent
#define LL     361    // number of degrees l
#define MM     362    // number of orders m
#define KLAT   361    // latitudes
#define NLON   722
#define TN     128    // TT*NLn rows of the Legendre GEMM
#define LPAD   368    // L padded (zero-filled) to a multiple of 16
#define MPAD   368    // M padded likewise
#define KW     736    // 2*MPAD : [re | im] concatenated K dim of the DFT GEMM
#define ROWS3  46208  // TN*KLAT rows of the DFT GEMM (= 32*1444, tiles exact)
#define XIROW  ((size_t)LL * MM * 2)   // xi stride between rows (t,n)

#define PHI_F   0.8824969025845955f   // exp(-DT/TIME_SCALE) = exp(-0.125)
#define FOURPI  12.566370614359172f

// workspace layout (floats)
#define A_ELEMS ((size_t)MM * 2 * LPAD * TN)     // 34,103,296
#define X_ELEMS ((size_t)KW * ROWS3)             // 34,009,088
// Wbuf follows: KW*NLON = 531,392

#define WMMA_F32(a, b, c) \
  __builtin_amdgcn_wmma_f32_16x16x4_f32(false, (a), false, (b), (short)0, (c), false, false)

// ---------------------------------------------------------------------------
// 1) AR(1) scan, emit-before-update. Thread owns one (m, comp, l) and walks
//    all 128 rows. Lanes contiguous in (m,comp) -> xi/coeff0 reads coalesced;
//    each thread writes a private contiguous 512B row block.
//    Block: 256 thr = comp(2) x m(16) x l(8). Grid: (23 m-chunks, 46 l-chunks)
//    Abuf[((m*2+comp)*LPAD + l)*TN + row]
// ---------------------------------------------------------------------------
__global__ __launch_bounds__(256) void ar1_scan(const float* __restrict__ sigma_n,
                                                const float* __restrict__ coeff0,
                                                const float* __restrict__ xi,
                                                float* __restrict__ Abuf) {
  const int tid  = threadIdx.x;
  const int comp = tid & 1;
  const int mo   = (tid >> 1) & 15;
  const int lo   = tid >> 5;
  const int m    = blockIdx.x * 16 + mo;     // [0, 368) -> guard 362
  const int l    = blockIdx.y * 8 + lo;      // [0, 368)
  if (m >= MM) return;

  float* Ap = Abuf + ((size_t)(m * 2 + comp) * LPAD + l) * TN;   // + row

  if (l >= LL) {                     // zero K-padding: v4f stores, 512B/thread
    const v4f z = {};
    #pragma unroll
    for (int r0 = 0; r0 < TN; r0 += 4) *(v4f*)(Ap + r0) = z;
    return;
  }

  const float s = sigma_n[l * MM + m];
  float st[NLn];
  #pragma unroll
  for (int n = 0; n < NLn; ++n)
    st[n] = coeff0[(((size_t)n * LL + l) * MM + m) * 2 + comp];

  const float* xp = xi + ((size_t)l * MM + m) * 2 + comp;        // + row*XIROW
  for (int t = 0; t < TT; ++t) {
    #pragma unroll
    for (int n = 0; n < NLn; ++n) {
      const int row = t * NLn + n;
      Ap[row] = st[n];                                   // emit BEFORE update
      st[n] = PHI_F * st[n] + s * xp[(size_t)row * XIROW];
    }
  }
}

// ---------------------------------------------------------------------------
// zero the m-padding K-slabs of Xbuf (c = 362..367 and 730..735), contiguous
// ---------------------------------------------------------------------------
__global__ __launch_bounds__(256) void zero_xpad(float* __restrict__ Xbuf) {
  const size_t id   = (size_t)blockIdx.x * blockDim.x + threadIdx.x;
  const size_t half = (size_t)6 * ROWS3;
  if (id >= 2 * half) return;
  const size_t off = (id < half) ? ((size_t)MM * ROWS3 + id)
                                 : ((size_t)(MPAD + MM) * ROWS3 + (id - half));
  Xbuf[off] = 0.0f;
}

// ---------------------------------------------------------------------------
// 3) DFT matrix W[c][j]: c<368 -> 4pi*w_m*cos(2pi*m*j/722),
//    c>=368 -> -4pi*w_m*sin(...). w_m=1 at DC/Nyquist (imag rows -> 0 there),
//    2 otherwise. Pads (m>=362) written as zero. Exact int phase reduction.
// ---------------------------------------------------------------------------
__global__ __launch_bounds__(256) void build_w(float* __restrict__ Wbuf) {
  const int id = blockIdx.x * blockDim.x + threadIdx.x;
  if (id >= KW * NLON) return;
  const int c = id / NLON;
  const int j = id - c * NLON;
  const int m = (c >= MPAD) ? (c - MPAD) : c;
  float v = 0.0f;
  if (m < MM) {
    const int ph = (m * j) % NLON;
    const float ang = (float)(6.283185307179586 * (double)ph / (double)NLON);
    const bool edge = (m == 0) || (m == MM - 1);
    if (c < MPAD) v = FOURPI * (edge ? 1.0f : 2.0f) * cosf(ang);
    else          v = edge ? 0.0f : (-FOURPI * 2.0f * sinf(ang));
  }
  Wbuf[id] = v;
}

// ---------------------------------------------------------------------------
// 2) Per-m Legendre GEMM, 32x32 macro-tile per wave (2x2 WMMA tiles).
//    C[row=tn][col=klat] = sum_l A_m[l][row] * pct[m][l][col]
//    Tiles: 362 m * 2 comp * 4 rowT2 * 12 colT2 = 34,752 waves = 4344 blocks.
//    coeffs are zero for l<m (tril) -> scalar K-loop starts at m&~3.
//    Ragged K-tail (l=360..363) peeled; B tail masked by value cndmask.
// ---------------------------------------------------------------------------
__global__ __launch_bounds__(256) void legendre_gemm(const float* __restrict__ pct,
                                                     const float* __restrict__ Abuf,
                                                     float* __restrict__ Xbuf) {
  const int lane = threadIdx.x & 31;
  const int tile = __builtin_amdgcn_readfirstlane(blockIdx.x * 8 + (threadIdx.x >> 5));
  const int m     = tile / 96;                    // 96 = 2*4*12 tiles per m
  int r           = tile - m * 96;
  const int comp  = r / 48;  r -= comp * 48;      // 48 = 4*12
  const int rowT2 = r / 12;
  const int colT2 = r - rowT2 * 12;
  const int kb0   = m & ~3;                       // scalar loop start

  const int hi    = lane >> 4;
  const int ln    = lane & 15;
  const int arow0 = rowT2 * 32 + ln;              // row halves: +0 / +16
  const int col0  = colT2 * 32 + ln;              // col halves: +0 / +16
  const int col1  = col0 + 16;
  const int cc0   = (col0 < KLAT) ? col0 : (KLAT - 1);   // clamp, no guard
  const int cc1   = (col1 < KLAT) ? col1 : (KLAT - 1);

  const float* Pb = pct + (size_t)m * LL * KLAT;
  const float* ap = Abuf + ((size_t)(m * 2 + comp) * LPAD + (kb0 + 2 * hi)) * TN + arow0;
  const float* bp = Pb + (size_t)(kb0 + 2 * hi) * KLAT;

  v8f acc00 = {}, acc01 = {}, acc10 = {}, acc11 = {};
  #pragma unroll 2
  for (int kb = kb0; kb < LL - 4; kb += 4) {      // last main kb = 356
    v2f a0;  a0.x = ap[0];        a0.y = ap[TN];
    v2f a1;  a1.x = ap[16];       a1.y = ap[TN + 16];
    v2f b0;  b0.x = bp[cc0];      b0.y = bp[KLAT + cc0];
    v2f b1;  b1.x = bp[cc1];      b1.y = bp[KLAT + cc1];
    acc00 = WMMA_F32(a0, b0, acc00);
    acc01 = WMMA_F32(a0, b1, acc01);
    acc10 = WMMA_F32(a1, b0, acc10);
    acc11 = WMMA_F32(a1, b1, acc11);
    ap += 4 * TN;  bp += 4 * KLAT;
  }
  { // tail kb = 360 : la = 360+2*hi; A pads are zero; B row clamped+masked
    const int la = 360 + 2 * hi;
    const size_t ro = (size_t)((la < LL) ? la : (LL - 1)) * KLAT;
    v2f a0;  a0.x = ap[0];   a0.y = ap[TN];
    v2f a1;  a1.x = ap[16];  a1.y = ap[TN + 16];
    const float bx0 = Pb[ro + cc0];
    const float bx1 = Pb[ro + cc1];
    v2f b0;  b0.x = (la < LL) ? bx0 : 0.0f;  b0.y = 0.0f;
    v2f b1;  b1.x = (la < LL) ? bx1 : 0.0f;  b1.y = 0.0f;
    acc00 = WMMA_F32(a0, b0, acc00);
    acc01 = WMMA_F32(a0, b1, acc01);
    acc10 = WMMA_F32(a1, b0, acc10);
    acc11 = WMMA_F32(a1, b1, acc11);
  }

  // stores (post-WMMA divergence OK); Xbuf[c][rowGlobal], lanes coalesced
  float* Xc = Xbuf + (size_t)(comp * MPAD + m) * ROWS3;
  #pragma unroll
  for (int rr = 0; rr < 8; ++rr) {                // D: VGPR rr = row rr / rr+8
    const int or0 = rowT2 * 32 + rr + 8 * hi;     // row half 0
    const int or1 = or0 + 16;                     // row half 1
    if (col0 < KLAT) {
      Xc[(size_t)or0 * KLAT + col0] = acc00[rr];
      Xc[(size_t)or1 * KLAT + col0] = acc10[rr];
    }
    if (col1 < KLAT) {
      Xc[(size_t)or0 * KLAT + col1] = acc01[rr];
      Xc[(size_t)or1 * KLAT + col1] = acc11[rr];
    }
  }
}

// ---------------------------------------------------------------------------
// 4) irfft as GEMM, 32x32 macro-tile per wave: out[row][j] = sum_c X[c][row]*W[c][j]
//    rows 46208 -> 1444 tiles, cols 722 -> 23 tiles; 33,212 waves -> 4152
//    blocks with a scalar whole-wave tail exit. No guards in the K-loop.
//    First col half (<=719) never clamps; only the second half does.
// ---------------------------------------------------------------------------
__global__ __launch_bounds__(256) void idft_gemm(const float* __restrict__ Xbuf,
                                                 const float* __restrict__ Wbuf,
                                                 float* __restrict__ out) {
  const int lane = threadIdx.x & 31;
  const int tile = __builtin_amdgcn_readfirstlane(blockIdx.x * 8 + (threadIdx.x >> 5));
  if (tile >= 1444 * 23) return;                  // scalar, whole-wave exit
  const int rowT2 = tile / 23;
  const int colT2 = tile - rowT2 * 23;

  const int hi    = lane >> 4;
  const int ln    = lane & 15;
  const int arow0 = rowT2 * 32 + ln;
  const int j0    = colT2 * 32 + ln;              // <= 719 : always valid
  const int j1    = j0 + 16;                      // may exceed 721
  const int j1c   = (j1 < NLON) ? j1 : (NLON - 1);

  const float* ap = Xbuf + (size_t)(2 * hi) * ROWS3 + arow0;
  const float* bp = Wbuf + (size_t)(2 * hi) * NLON;

  v8f acc00 = {}, acc01 = {}, acc10 = {}, acc11 = {};
  #pragma unroll 2
  for (int kb = 0; kb < KW; kb += 4) {
    v2f a0;  a0.x = ap[0];      a0.y = ap[ROWS3];
    v2f a1;  a1.x = ap[16];     a1.y = ap[ROWS3 + 16];
    v2f b0;  b0.x = bp[j0];     b0.y = bp[NLON + j0];
    v2f b1;  b1.x = bp[j1c];    b1.y = bp[NLON + j1c];
    acc00 = WMMA_F32(a0, b0, acc00);
    acc01 = WMMA_F32(a0, b1, acc01);
    acc10 = WMMA_F32(a1, b0, acc10);
    acc11 = WMMA_F32(a1, b1, acc11);
    ap += (size_t)4 * ROWS3;  bp += 4 * NLON;
  }

  #pragma unroll
  for (int rr = 0; rr < 8; ++rr) {
    const int or0 = rowT2 * 32 + rr + 8 * hi;     // == (t*16+n)*361 + k
    const int or1 = or0 + 16;
    out[(size_t)or0 * NLON + j0] = acc00[rr];
    out[(size_t)or1 * NLON + j0] = acc10[rr];
    if (j1 < NLON) {
      out[(size_t)or0 * NLON + j1] = acc01[rr];
      out[(size_t)or1 * NLON + j1] = acc11[rr];
    }
  }
}

// ---------------------------------------------------------------------------
extern "C" void kernel_launch(void* const* d_in, const int* in_sizes, int n_in,
                              void* d_out, int out_size, void* d_ws, size_t ws_size,
                              hipStream_t stream) {
  // setup_inputs order: x(int,unused), sigma_n, coeff0, xi, pct
  const float* sigma_n = (const float*)d_in[1];
  const float* coeff0  = (const float*)d_in[2];
  const float* xi      = (const float*)d_in[3];
  const float* pct     = (const float*)d_in[4];
  float* out  = (float*)d_out;
  float* Abuf = (float*)d_ws;
  float* Xbuf = Abuf + A_ELEMS;
  float* Wbuf = Xbuf + X_ELEMS;

  ar1_scan<<<dim3(23, 46), 256, 0, stream>>>(sigma_n, coeff0, xi, Abuf);
  {
    const int total = 12 * ROWS3;                 // 554,496
    zero_xpad<<<(total + 255) / 256, 256, 0, stream>>>(Xbuf);
  }
  {
    const int total = KW * NLON;                  // 531,392
    build_w<<<(total + 255) / 256, 256, 0, stream>>>(Wbuf);
  }
  legendre_gemm<<<4344, 256, 0, stream>>>(pct, Abuf, Xbuf);  // 362*96 /8
  idft_gemm   <<<4152, 256, 0, stream>>>(Xbuf, Wbuf, out);   // 1444*23 /8 (pad)
}